// TransformerCell_68856915689599
// MI455X (gfx1250) — compile-verified
//
#include <hip/hip_runtime.h>
#include <math.h>

typedef __attribute__((ext_vector_type(2))) float v2f;
typedef __attribute__((ext_vector_type(4))) float v4f;
typedef __attribute__((ext_vector_type(8))) float v8f;

#define BATCH 16
#define TLEN  8192
#define DIM   1024
#define TP1   8193
#define NBLK  64            // k1 blocks per batch
#define WAVES 8
#define WS_STRIDE 1028      // floats per partial: 1024 acc + M + L + pad (16B aligned)

__device__ __forceinline__ v4f fma4(v4f a, float s, v4f acc) {
    acc.x = fmaf(a.x, s, acc.x);
    acc.y = fmaf(a.y, s, acc.y);
    acc.z = fmaf(a.z, s, acc.z);
    acc.w = fmaf(a.w, s, acc.w);
    return acc;
}

// Kernel 1: per wave, one 16-key tile: copy rows (NT store), WMMA f32 scores,
// tile softmax, weighted-V accumulation; block-merge 8 waves -> 1 partial in ws.
__global__ __launch_bounds__(256)
void attn_partial_kernel(const float* __restrict__ token,
                         const float* __restrict__ prev,
                         float* __restrict__ out_prev,
                         float* __restrict__ ws)
{
    const int b    = blockIdx.y;
    const int lane = threadIdx.x & 31;
    const int wave = threadIdx.x >> 5;
    const int tile = blockIdx.x * WAVES + wave;    // 0..511
    const int t0   = tile * 16;

    const float* __restrict__ kbase = prev     + ((size_t)b * TLEN + t0) * DIM;
    float*       __restrict__ obase = out_prev + ((size_t)b * TP1  + t0) * DIM;
    const float* __restrict__ q     = token    + (size_t)b * DIM;

    // ---- Phase A: stream-copy 16 rows, coalesced b128; NT stores (copy is never re-read) ----
    #pragma unroll
    for (int m = 0; m < 16; ++m) {
        const v4f* src = (const v4f*)(kbase + m * DIM);
        v4f*       dst = (v4f*)(obase + m * DIM);
        #pragma unroll
        for (int c = 0; c < 8; ++c) {
            v4f v = src[c * 32 + lane];
            __builtin_nontemporal_store(v, dst + c * 32 + lane);
        }
    }

    // ---- Phase B: scores for 16 keys via V_WMMA_F32_16X16X4_F32, C-chained over D ----
    // A layout (16x4 f32): lane m holds row m; VGPR0 = K{0|2}, VGPR1 = K{1|3} per lane-half.
    // => per-lane b64 load at col d + 2*(lane>>4). B = broadcast q slice, same addressing.
    const int half = lane >> 4;
    const int row  = lane & 15;
    const float* arow = kbase + row * DIM + 2 * half;
    const float* brow = q + 2 * half;
    v8f c8 = {0.f, 0.f, 0.f, 0.f, 0.f, 0.f, 0.f, 0.f};
    for (int d = 0; d < DIM; d += 4) {
        v2f a  = *(const v2f*)(arow + d);
        v2f bb = *(const v2f*)(brow + d);
        c8 = __builtin_amdgcn_wmma_f32_16x16x4_f32(false, a, false, bb,
                                                   (short)0, c8, false, false);
    }
    // c8[r]: lanes 0-15 = score(key t0+r), lanes 16-31 = score(key t0+8+r)

    // ---- Phase C: tile softmax stats ----
    const float inv_sqrt_d = 0.03125f;   // 1/sqrt(1024)
    float s0[8];
    float mloc = -3.4e38f;
    #pragma unroll
    for (int r = 0; r < 8; ++r) { s0[r] = c8[r] * inv_sqrt_d; mloc = fmaxf(mloc, s0[r]); }
    float mt = fmaxf(mloc, __shfl_xor(mloc, 16, 32));   // tile max (all lanes)
    float p[8];
    float lh = 0.f;
    #pragma unroll
    for (int r = 0; r < 8; ++r) { p[r] = __expf(s0[r] - mt); lh += p[r]; }
    float lt = lh + __shfl_xor(lh, 16, 32);             // tile sum (all lanes)

    // ---- Phase D: acc[d] = sum_m p_m * K[m][d]  (cache-hot reloads, coalesced) ----
    v4f acc[8];
    #pragma unroll
    for (int c = 0; c < 8; ++c) acc[c] = (v4f){0.f, 0.f, 0.f, 0.f};
    #pragma unroll
    for (int m = 0; m < 16; ++m) {
        float pm = __shfl(p[m & 7], (m < 8) ? 0 : 16, 32);
        const v4f* src = (const v4f*)(kbase + m * DIM);
        #pragma unroll
        for (int c = 0; c < 8; ++c) acc[c] = fma4(src[c * 32 + lane], pm, acc[c]);
    }

    // ---- Phase E: merge 8 waves -> one block partial ----
    __shared__ v4f   lds_acc[WAVES][256];
    __shared__ float lds_ml[WAVES][2];
    if (lane == 0) { lds_ml[wave][0] = mt; lds_ml[wave][1] = lt; }
    __syncthreads();
    float Mb = lds_ml[0][0];
    #pragma unroll
    for (int w = 1; w < WAVES; ++w) Mb = fmaxf(Mb, lds_ml[w][0]);
    float sc = __expf(mt - Mb);
    #pragma unroll
    for (int c = 0; c < 8; ++c) {
        lds_acc[wave][c * 32 + lane] = acc[c] * sc;
    }
    if (lane == 0) lds_ml[wave][1] = lt * sc;
    __syncthreads();

    const int t = threadIdx.x;               // 0..255 -> dims 4t..4t+3
    v4f sum = lds_acc[0][t];
    #pragma unroll
    for (int w = 1; w < WAVES; ++w) sum += lds_acc[w][t];
    float Lb = 0.f;
    #pragma unroll
    for (int w = 0; w < WAVES; ++w) Lb += lds_ml[w][1];

    float* wsp = ws + (size_t)(b * NBLK + blockIdx.x) * WS_STRIDE;
    ((v4f*)wsp)[t] = sum;
    if (t == 0) { wsp[1024] = Mb; wsp[1025] = Lb; }
}

// Kernel 2: per batch, merge 64 partials + self-attention term; write result and
// append the query row to next_previous.
__global__ __launch_bounds__(256)
void attn_finalize_kernel(const float* __restrict__ token,
                          const float* __restrict__ ws,
                          float* __restrict__ result,
                          float* __restrict__ out_prev)
{
    const int b = blockIdx.x;
    const int t = threadIdx.x;
    const float* __restrict__ q = token + (size_t)b * DIM;

    __shared__ float red[256];
    v4f q4 = ((const v4f*)q)[t];
    red[t] = q4.x * q4.x + q4.y * q4.y + q4.z * q4.z + q4.w * q4.w;
    __syncthreads();
    for (int off = 128; off > 0; off >>= 1) {
        if (t < off) red[t] += red[t + off];
        __syncthreads();
    }
    const float s_self = red[0] * 0.03125f;

    float M = s_self;
    for (int c = 0; c < NBLK; ++c)
        M = fmaxf(M, ws[(size_t)(b * NBLK + c) * WS_STRIDE + 1024]);

    float e_self = __expf(s_self - M);
    float L = e_self;
    v4f accv = q4 * e_self;
    for (int c = 0; c < NBLK; ++c) {
        const float* wsp = ws + (size_t)(b * NBLK + c) * WS_STRIDE;
        float e = __expf(wsp[1024] - M);
        L += wsp[1025] * e;
        v4f a4 = ((const v4f*)wsp)[t];
        accv = fma4(a4, e, accv);
    }
    float rinv = 1.0f / L;
    accv *= rinv;
    ((v4f*)(result + (size_t)b * DIM))[t] = accv;

    // append query as last cache row
    v4f* last = (v4f*)(out_prev + ((size_t)b * TP1 + TLEN) * DIM);
    __builtin_nontemporal_store(q4, last + t);
}

extern "C" void kernel_launch(void* const* d_in, const int* in_sizes, int n_in,
                              void* d_out, int out_size, void* d_ws, size_t ws_size,
                              hipStream_t stream)
{
    const float* token = (const float*)d_in[0];   // (16, 1024)
    const float* prev  = (const float*)d_in[1];   // (16, 8192, 1024)
    float* result   = (float*)d_out;                          // (16, 1024)
    float* out_prev = (float*)d_out + (size_t)BATCH * DIM;    // (16, 8193, 1024)
    float* ws = (float*)d_ws;   // needs 16*64*1028*4 B ~= 4.2 MB

    attn_partial_kernel<<<dim3(NBLK, BATCH), 256, 0, stream>>>(token, prev, out_prev, ws);
    attn_finalize_kernel<<<dim3(BATCH), 256, 0, stream>>>(token, ws, result, out_prev);
}